// ModelNew_11888469476211
// MI455X (gfx1250) — compile-verified
//
#include <hip/hip_runtime.h>
#include <hip/hip_bf16.h>
#include <stdint.h>

// ---------------------------------------------------------------------------
// Reference collapses to a constant:
//   y = clip(min(gn(conv(x)), 0.25), 0.25, 1.0)  ==  0.25 everywhere (finite in)
// Optimal MI455X kernel = stream 244 MB of 0.25f at HBM write bandwidth
// (~10.5 us @ 23.3 TB/s). No FLOPs, no input reads.
//
// Bulk path: gfx1250 async LDS->global store engine
// (GLOBAL_STORE_ASYNC_FROM_LDS_B128, ASYNCcnt). 512 B constant staged in LDS
// once per block; each wave then owns contiguous 4 KB spans and issues 8
// async B128 stores per span using immediate IOFFSETs (0..3584), i.e. one
// 64-bit address computation per 4 KB written. Fallback: nontemporal stores.
// ---------------------------------------------------------------------------

#ifndef __has_builtin
#define __has_builtin(x) 0
#endif

#define AS1 __attribute__((address_space(1)))
#define AS3 __attribute__((address_space(3)))

#if defined(__gfx1250__) && __has_builtin(__builtin_amdgcn_global_store_async_from_lds_b128)
#define USE_ASYNC_LDS_STORE 1
#else
#define USE_ASYNC_LDS_STORE 0
#endif

typedef float v4f __attribute__((ext_vector_type(4)));
typedef int   v4i __attribute__((ext_vector_type(4)));

__global__ __launch_bounds__(256) void fill_const_kernel(float* __restrict__ out,
                                                         long long nvec4,
                                                         long long total,
                                                         float value)
{
    const long long tid    = (long long)blockIdx.x * blockDim.x + threadIdx.x;
    const long long stride = (long long)gridDim.x * blockDim.x;

#if USE_ASYNC_LDS_STORE
    // 512 B staging buffer: lane L of every wave owns stage[4L .. 4L+3].
    __shared__ float stage[128];
    if (threadIdx.x < 128) stage[threadIdx.x] = value;
    __syncthreads();  // DS writes visible before the async engine reads LDS

    const int lane = threadIdx.x & 31;
    // AS3 pointer = low 32 bits of the flat LDS address (ISA aperture rule).
    AS3 v4i* ldsSrc = (AS3 v4i*)(uint32_t)(uintptr_t)&stage[lane * 4];

    // ---- Bulk: contiguous 4 KB spans per wave, 8 stores @ immediate offsets.
    const long long wid    = tid >> 5;       // global wave id
    const long long nwaves = stride >> 5;
    const long long SPAN   = 32 * 8;         // vec4s per span (4 KB)
    const long long nspan  = nvec4 / SPAN;

    for (long long s = wid; s < nspan; s += nwaves) {
        AS1 v4i* dst = (AS1 v4i*)(uintptr_t)(out + (s * SPAN + lane) * 4);
        // 8 coalesced 512 B wave-writes; IOFFSET must be an ICE -> literals.
        __builtin_amdgcn_global_store_async_from_lds_b128(dst, ldsSrc,    0, 0);
        __builtin_amdgcn_global_store_async_from_lds_b128(dst, ldsSrc,  512, 0);
        __builtin_amdgcn_global_store_async_from_lds_b128(dst, ldsSrc, 1024, 0);
        __builtin_amdgcn_global_store_async_from_lds_b128(dst, ldsSrc, 1536, 0);
        __builtin_amdgcn_global_store_async_from_lds_b128(dst, ldsSrc, 2048, 0);
        __builtin_amdgcn_global_store_async_from_lds_b128(dst, ldsSrc, 2560, 0);
        __builtin_amdgcn_global_store_async_from_lds_b128(dst, ldsSrc, 3072, 0);
        __builtin_amdgcn_global_store_async_from_lds_b128(dst, ldsSrc, 3584, 0);
        // ASYNCcnt (max 63 outstanding) throttles issue in hardware.
    }

    // ---- Remainder vec4s (none for 61,011,968 elements, but stay generic).
    for (long long i = nspan * SPAN + tid; i < nvec4; i += stride) {
        AS1 v4i* dst = (AS1 v4i*)(uintptr_t)(out + 4 * i);
        __builtin_amdgcn_global_store_async_from_lds_b128(dst, ldsSrc, 0, 0);
    }

#if __has_builtin(__builtin_amdgcn_s_wait_asynccnt)
    __builtin_amdgcn_s_wait_asynccnt(0);   // explicit drain (endpgm also drains)
#endif
#else
    // Fallback: 128-bit nontemporal streaming stores (still bandwidth-optimal).
    const v4f v = {value, value, value, value};
    for (long long i = tid; i < nvec4; i += stride) {
        __builtin_nontemporal_store(v, (v4f*)(out + 4 * i));
    }
#endif

    // Scalar tail (out_size is divisible by 4 here, but stay generic).
    if (tid == 0) {
        for (long long j = nvec4 * 4; j < total; ++j) out[j] = value;
    }
}

extern "C" void kernel_launch(void* const* d_in, const int* in_sizes, int n_in,
                              void* d_out, int out_size, void* d_ws, size_t ws_size,
                              hipStream_t stream)
{
    (void)d_in; (void)in_sizes; (void)n_in; (void)d_ws; (void)ws_size;

    float* out = (float*)d_out;            // output dtype: float32
    const long long total = (long long)out_size;   // 8*32*62^3 = 61,011,968
    const long long nvec4 = total / 4;

    const int threads = 256;               // 8 wave32 waves per block
    const int blocks  = 3072;              // 24576 waves, grid-stride over spans

    fill_const_kernel<<<blocks, threads, 0, stream>>>(out, nvec4, total, 0.25f);
}